// TensorNetwork_56461640073794
// MI455X (gfx1250) — compile-verified
//
#include <hip/hip_runtime.h>
#include <hip/hip_bf16.h>
#include <math.h>

#define NNODE 4096
#define NEDGE 32768
#define HC    128
#define RC    32
#define NLAY  2

typedef __attribute__((ext_vector_type(2))) float v2f;
typedef __attribute__((ext_vector_type(8))) float v8f;

// ---------------------------------------------------------------------------
// Strided fp32 GEMM on the WMMA pipe: Out = epilogue(A @ B + bias)
//   A[m,k]   = A[m*lda + k*KA + z*za]      (KA compile-time: 1 fast path, 3/6 cmix)
//   B[k,n]   = B[k*Nc + n]                 (row-major [K,Nc])
//   Out[m,n] = Out[m*ldo + n*ko + z*zo]
// Epilogue order: +bias -> silu (if act) -> *rowscale[m] (if given)
// One wave computes a 64x16 tile: 4 independent V_WMMA_F32_16X16X4_F32 chains
// share each B fragment; A fragments are b64 vector loads when KA==1.
// ---------------------------------------------------------------------------
template <int KA>
__global__ void k_gemm_wmma(const float* __restrict__ A, int lda,
                            const float* __restrict__ B,
                            const float* __restrict__ bias,
                            const float* __restrict__ rowscale,
                            float* __restrict__ Out, int ldo, int ko,
                            int M, int Nc, int K, int za, int zo, int act)
{
    const int lane    = threadIdx.x & 31;
    const int wave    = threadIdx.x >> 5;
    const int tiles_n = Nc >> 4;
    const int tiles_m = M >> 6;                     // 64 rows per wave
    const int tile    = blockIdx.x * (blockDim.x >> 5) + wave;
    if (tile >= tiles_m * tiles_n) return;          // wave-uniform: EXEC stays all-ones
    const int tm = tile / tiles_n;
    const int tn = tile - tm * tiles_n;
    const size_t zA = (size_t)blockIdx.z * (size_t)za;
    const size_t zO = (size_t)blockIdx.z * (size_t)zo;
    const int half = lane >> 4;                     // 0: lanes 0-15, 1: lanes 16-31
    const int l16  = lane & 15;
    const unsigned n0 = tn * 16 + l16;

    // Row base for this lane within each 16-row sub-tile (s = 0..3)
    const float* Ap = A + (size_t)(tm * 64 + l16) * (size_t)lda + zA;
    const unsigned subStride = 16u * (unsigned)lda; // elements between sub-tiles

    const v8f vzero = {0.f, 0.f, 0.f, 0.f, 0.f, 0.f, 0.f, 0.f};
    v8f acc0 = vzero, acc1 = vzero, acc2 = vzero, acc3 = vzero;

    for (int k0 = 0; k0 < K; k0 += 4) {
        const unsigned kk = (unsigned)(k0 + 2 * half);  // ISA 16x4 f32 A layout
        v2f b;
        b.x = B[kk * (unsigned)Nc + n0];
        b.y = B[(kk + 1u) * (unsigned)Nc + n0];
        v2f a0, a1, a2, a3;
        if (KA == 1) {
            a0 = *(const v2f*)(Ap + 0u * subStride + kk);
            a1 = *(const v2f*)(Ap + 1u * subStride + kk);
            a2 = *(const v2f*)(Ap + 2u * subStride + kk);
            a3 = *(const v2f*)(Ap + 3u * subStride + kk);
        } else {
            const unsigned o0 = kk * (unsigned)KA, o1 = (kk + 1u) * (unsigned)KA;
            a0.x = Ap[0u * subStride + o0]; a0.y = Ap[0u * subStride + o1];
            a1.x = Ap[1u * subStride + o0]; a1.y = Ap[1u * subStride + o1];
            a2.x = Ap[2u * subStride + o0]; a2.y = Ap[2u * subStride + o1];
            a3.x = Ap[3u * subStride + o0]; a3.y = Ap[3u * subStride + o1];
        }
        acc0 = __builtin_amdgcn_wmma_f32_16x16x4_f32(false, a0, false, b, (short)0, acc0, false, false);
        acc1 = __builtin_amdgcn_wmma_f32_16x16x4_f32(false, a1, false, b, (short)0, acc1, false, false);
        acc2 = __builtin_amdgcn_wmma_f32_16x16x4_f32(false, a2, false, b, (short)0, acc2, false, false);
        acc3 = __builtin_amdgcn_wmma_f32_16x16x4_f32(false, a3, false, b, (short)0, acc3, false, false);
    }

    const float bv = bias ? bias[n0] : 0.0f;
    v8f accs[4] = {acc0, acc1, acc2, acc3};
#pragma unroll
    for (int s = 0; s < 4; ++s) {
#pragma unroll
        for (int v = 0; v < 8; ++v) {
            const int m = tm * 64 + s * 16 + v + 8 * half;  // C/D: VGPR v -> rows v / v+8
            float x = accs[s][v] + bv;
            if (act) x = x / (1.0f + expf(-x));             // silu
            if (rowscale) x *= rowscale[m];
            Out[(size_t)m * (size_t)ldo + (size_t)n0 * (size_t)ko + zO] = x;
        }
    }
}

// ---------------------------------------------------------------------------
// Elementwise / scatter kernels
// ---------------------------------------------------------------------------
__global__ void k_zero(float* __restrict__ p, size_t n)
{
    size_t i = (size_t)blockIdx.x * blockDim.x + threadIdx.x;
    if (i < n) p[i] = 0.0f;
}

// C[e] = cosine cutoff; evn[e] = masked-normalized edge vector
__global__ void k_edge_prep(const float* __restrict__ ew, const float* __restrict__ evec,
                            const int* __restrict__ ei, float* __restrict__ C,
                            float* __restrict__ evn)
{
    int e = blockIdx.x * blockDim.x + threadIdx.x;
    if (e >= NEDGE) return;
    float w = ew[e];
    float c = 0.5f * (cosf(w * (3.14159265358979323846f / 5.0f)) + 1.0f);
    C[e] = (w < 5.0f) ? c : 0.0f;
    int s = ei[e], d = ei[NEDGE + e];
    float x = evec[e * 3 + 0], y = evec[e * 3 + 1], zc = evec[e * 3 + 2];
    if (s != d) {
        float nrm = sqrtf(x * x + y * y + zc * zc);
        float inv = 1.0f / fmaxf(nrm, 1e-12f);
        x *= inv; y *= inv; zc *= inv;
    }
    evn[e * 3 + 0] = x; evn[e * 3 + 1] = y; evn[e * 3 + 2] = zc;
}

// Zcat[e, 0:128] = emb[z[dst]], Zcat[e, 128:256] = emb[z[src]]
__global__ void k_zcat(const int* __restrict__ z, const int* __restrict__ ei,
                       const float* __restrict__ emb, float* __restrict__ Zcat)
{
    size_t t = (size_t)blockIdx.x * blockDim.x + threadIdx.x;
    if (t >= (size_t)NEDGE * 2 * HC) return;
    int e = (int)(t >> 8);
    int k = (int)(t & 255);
    int node = (k < HC) ? ei[NEDGE + e] : ei[e];
    Zcat[t] = emb[z[node] * HC + (k & (HC - 1))];
}

// Scatter for TNE: per (e,h) 10 atomic f32 adds into compact I/A/S accumulators.
__global__ void k_tne_scatter(const float* __restrict__ Zij, const float* __restrict__ W123,
                              const float* __restrict__ evn, const int* __restrict__ ei,
                              float* __restrict__ iS, float* __restrict__ aV,
                              float* __restrict__ sV)
{
    size_t t = (size_t)blockIdx.x * blockDim.x + threadIdx.x;
    if (t >= (size_t)NEDGE * HC) return;
    const int e = (int)(t >> 7), h = (int)(t & (HC - 1));
    const size_t EH = (size_t)NEDGE * HC;
    float zij = Zij[t];
    float p1 = zij * W123[t];
    float p2 = zij * W123[EH + t];
    float p3 = zij * W123[2 * EH + t];
    int n = ei[NEDGE + e];                          // dst segments
    size_t idx = (size_t)n * HC + h;
    atomicAdd(&iS[idx], p1);
    float vx = evn[e * 3 + 0], vy = evn[e * 3 + 1], vz = evn[e * 3 + 2];
    atomicAdd(&aV[idx * 3 + 0], p2 * vx);           // skew() is linear: scatter the vector
    atomicAdd(&aV[idx * 3 + 1], p2 * vy);
    atomicAdd(&aV[idx * 3 + 2], p2 * vz);
    float q = (vx * vx + vy * vy + vz * vz) * (1.0f / 3.0f);
    atomicAdd(&sV[idx * 6 + 0], p3 * (vx * vx - q));
    atomicAdd(&sV[idx * 6 + 1], p3 * (vx * vy));
    atomicAdd(&sV[idx * 6 + 2], p3 * (vx * vz));
    atomicAdd(&sV[idx * 6 + 3], p3 * (vy * vy - q));
    atomicAdd(&sV[idx * 6 + 4], p3 * (vy * vz));
    atomicAdd(&sV[idx * 6 + 5], p3 * (vz * vz - q));
}

// tensor_norm from compact rep (I/A/S are Frobenius-orthogonal), then LayerNorm over H.
__global__ void k_norm_ln(const float* __restrict__ iS, const float* __restrict__ aV,
                          const float* __restrict__ sV, const float* __restrict__ g,
                          const float* __restrict__ b, float* __restrict__ outln)
{
    const int n = blockIdx.x, h = threadIdx.x;      // blockDim = 128
    const size_t idx = (size_t)n * HC + h;
    float ic = iS[idx];
    const float* a = aV + idx * 3;
    const float* s = sV + idx * 6;
    float tn = 3.0f * ic * ic
             + 2.0f * (a[0] * a[0] + a[1] * a[1] + a[2] * a[2])
             + s[0] * s[0] + s[3] * s[3] + s[5] * s[5]
             + 2.0f * (s[1] * s[1] + s[2] * s[2] + s[4] * s[4]);
    __shared__ float red[HC];
    red[h] = tn; __syncthreads();
    for (int off = HC / 2; off > 0; off >>= 1) {
        if (h < off) red[h] += red[h + off];
        __syncthreads();
    }
    float mean = red[0] * (1.0f / HC);
    __syncthreads();
    float d = tn - mean;
    red[h] = d * d; __syncthreads();
    for (int off = HC / 2; off > 0; off >>= 1) {
        if (h < off) red[h] += red[h + off];
        __syncthreads();
    }
    float var = red[0] * (1.0f / HC);
    outln[idx] = d * rsqrtf(var + 1e-5f) * g[h] + b[h];
}

__device__ inline void assemble9(float ic, float a0, float a1, float a2,
                                 float s0, float s1, float s2, float s3, float s4, float s5,
                                 float* T)
{
    T[0] = ic + s0;  T[1] = -a2 + s1; T[2] = a1 + s2;
    T[3] = a2 + s1;  T[4] = ic + s3;  T[5] = -a0 + s4;
    T[6] = -a1 + s2; T[7] = a0 + s4;  T[8] = ic + s5;
}

// X = f0*I + f1*A + f2*S  (TNE output)
__global__ void k_assemble_X(const float* __restrict__ iC, const float* __restrict__ aC,
                             const float* __restrict__ sC, const float* __restrict__ fsc,
                             float* __restrict__ X)
{
    size_t t = (size_t)blockIdx.x * blockDim.x + threadIdx.x;
    if (t >= (size_t)NNODE * HC) return;
    int n = (int)(t >> 7), h = (int)(t & (HC - 1));
    size_t fb = (size_t)n * 3 * HC + (size_t)h * 3;
    float f0 = fsc[fb], f1 = fsc[fb + 1], f2 = fsc[fb + 2];
    assemble9(f0 * iC[t],
              f1 * aC[t * 3 + 0], f1 * aC[t * 3 + 1], f1 * aC[t * 3 + 2],
              f2 * sC[t * 6 + 0], f2 * sC[t * 6 + 1], f2 * sC[t * 6 + 2],
              f2 * sC[t * 6 + 3], f2 * sC[t * 6 + 4], f2 * sC[t * 6 + 5],
              X + t * 9);
}

// X <- X/(||X||^2+1) (written back), and compact decomposition of the scaled X.
__global__ void k_decomp(float* __restrict__ X, float* __restrict__ iC,
                         float* __restrict__ aC, float* __restrict__ sC)
{
    size_t t = (size_t)blockIdx.x * blockDim.x + threadIdx.x;
    if (t >= (size_t)NNODE * HC) return;
    float T[9];
    float tn = 0.f;
#pragma unroll
    for (int i = 0; i < 9; ++i) { T[i] = X[t * 9 + i]; tn += T[i] * T[i]; }
    float sc = 1.0f / (tn + 1.0f);
#pragma unroll
    for (int i = 0; i < 9; ++i) { T[i] *= sc; X[t * 9 + i] = T[i]; }
    float ic = (T[0] + T[4] + T[8]) * (1.0f / 3.0f);
    iC[t] = ic;
    aC[t * 3 + 0] = 0.5f * (T[7] - T[5]);
    aC[t * 3 + 1] = 0.5f * (T[2] - T[6]);
    aC[t * 3 + 2] = 0.5f * (T[3] - T[1]);
    sC[t * 6 + 0] = T[0] - ic;
    sC[t * 6 + 1] = 0.5f * (T[1] + T[3]);
    sC[t * 6 + 2] = 0.5f * (T[2] + T[6]);
    sC[t * 6 + 3] = T[4] - ic;
    sC[t * 6 + 4] = 0.5f * (T[5] + T[7]);
    sC[t * 6 + 5] = T[8] - ic;
}

// Y = I + A + S (after cmix), full 3x3 per (n,h)
__global__ void k_assembleY(const float* __restrict__ iC, const float* __restrict__ aC,
                            const float* __restrict__ sC, float* __restrict__ Y)
{
    size_t t = (size_t)blockIdx.x * blockDim.x + threadIdx.x;
    if (t >= (size_t)NNODE * HC) return;
    assemble9(iC[t],
              aC[t * 3 + 0], aC[t * 3 + 1], aC[t * 3 + 2],
              sC[t * 6 + 0], sC[t * 6 + 1], sC[t * 6 + 2],
              sC[t * 6 + 3], sC[t * 6 + 4], sC[t * 6 + 5],
              Y + t * 9);
}

// Message scatter: per (e,h) gather compact I/A/S of src node, scale by ea channels,
// 10 atomics into dst accumulators.
__global__ void k_msg_scatter(const float* __restrict__ ea, const int* __restrict__ ei,
                              const float* __restrict__ iC, const float* __restrict__ aC,
                              const float* __restrict__ sC, float* __restrict__ mi,
                              float* __restrict__ mav, float* __restrict__ msv)
{
    size_t t = (size_t)blockIdx.x * blockDim.x + threadIdx.x;
    if (t >= (size_t)NEDGE * HC) return;
    const int e = (int)(t >> 7), h = (int)(t & (HC - 1));
    size_t eb = (size_t)e * 3 * HC + (size_t)h * 3;
    float e0 = ea[eb], e1 = ea[eb + 1], e2 = ea[eb + 2];
    int ns = ei[e];                                 // src (gather)
    int nd = ei[NEDGE + e];                         // dst (scatter)
    size_t si = (size_t)ns * HC + h, di = (size_t)nd * HC + h;
    atomicAdd(&mi[di], e0 * iC[si]);
#pragma unroll
    for (int c = 0; c < 3; ++c) atomicAdd(&mav[di * 3 + c], e1 * aC[si * 3 + c]);
#pragma unroll
    for (int c = 0; c < 6; ++c) atomicAdd(&msv[di * 6 + c], e2 * sC[si * 6 + c]);
}

// AB = msg@Y + Y@msg; compact decomposition of AB scaled by 1/(||AB||^2+1)
__global__ void k_AB(const float* __restrict__ mi, const float* __restrict__ mav,
                     const float* __restrict__ msv, const float* __restrict__ Y,
                     float* __restrict__ iC, float* __restrict__ aC, float* __restrict__ sC)
{
    size_t t = (size_t)blockIdx.x * blockDim.x + threadIdx.x;
    if (t >= (size_t)NNODE * HC) return;
    float M9[9], Y9[9], AB[9];
    assemble9(mi[t], mav[t * 3 + 0], mav[t * 3 + 1], mav[t * 3 + 2],
              msv[t * 6 + 0], msv[t * 6 + 1], msv[t * 6 + 2],
              msv[t * 6 + 3], msv[t * 6 + 4], msv[t * 6 + 5], M9);
#pragma unroll
    for (int i = 0; i < 9; ++i) Y9[i] = Y[t * 9 + i];
    float tn = 0.f;
#pragma unroll
    for (int i = 0; i < 3; ++i)
#pragma unroll
        for (int j = 0; j < 3; ++j) {
            float acc = 0.f;
#pragma unroll
            for (int k = 0; k < 3; ++k)
                acc += M9[i * 3 + k] * Y9[k * 3 + j] + Y9[i * 3 + k] * M9[k * 3 + j];
            AB[i * 3 + j] = acc;
            tn += acc * acc;
        }
    float inv = 1.0f / (tn + 1.0f);
    float ic = (AB[0] + AB[4] + AB[8]) * (1.0f / 3.0f);
    iC[t] = ic * inv;
    aC[t * 3 + 0] = 0.5f * (AB[7] - AB[5]) * inv;
    aC[t * 3 + 1] = 0.5f * (AB[2] - AB[6]) * inv;
    aC[t * 3 + 2] = 0.5f * (AB[3] - AB[1]) * inv;
    sC[t * 6 + 0] = (AB[0] - ic) * inv;
    sC[t * 6 + 1] = 0.5f * (AB[1] + AB[3]) * inv;
    sC[t * 6 + 2] = 0.5f * (AB[2] + AB[6]) * inv;
    sC[t * 6 + 3] = (AB[4] - ic) * inv;
    sC[t * 6 + 4] = 0.5f * (AB[5] + AB[7]) * inv;
    sC[t * 6 + 5] = (AB[8] - ic) * inv;
}

// dX = I+A+S; dX += dX@dX; X += dX
__global__ void k_update(const float* __restrict__ iC, const float* __restrict__ aC,
                         const float* __restrict__ sC, float* __restrict__ X)
{
    size_t t = (size_t)blockIdx.x * blockDim.x + threadIdx.x;
    if (t >= (size_t)NNODE * HC) return;
    float D[9], DD[9];
    assemble9(iC[t], aC[t * 3 + 0], aC[t * 3 + 1], aC[t * 3 + 2],
              sC[t * 6 + 0], sC[t * 6 + 1], sC[t * 6 + 2],
              sC[t * 6 + 3], sC[t * 6 + 4], sC[t * 6 + 5], D);
#pragma unroll
    for (int i = 0; i < 3; ++i)
#pragma unroll
        for (int j = 0; j < 3; ++j) {
            float acc = 0.f;
#pragma unroll
            for (int k = 0; k < 3; ++k) acc += D[i * 3 + k] * D[k * 3 + j];
            DD[i * 3 + j] = acc;
        }
#pragma unroll
    for (int i = 0; i < 9; ++i) X[t * 9 + i] += D[i] + DD[i];
}

// ---------------------------------------------------------------------------
extern "C" void kernel_launch(void* const* d_in, const int* in_sizes, int n_in,
                              void* d_out, int out_size, void* d_ws, size_t ws_size,
                              hipStream_t stream)
{
    const int*   z      = (const int*)  d_in[0];
    const int*   ei     = (const int*)  d_in[1];
    const float* ew     = (const float*)d_in[2];
    const float* evec   = (const float*)d_in[3];
    const float* eattr  = (const float*)d_in[4];
    const float* emb    = (const float*)d_in[5];
    const float* dp1_w  = (const float*)d_in[6];
    const float* dp1_b  = (const float*)d_in[7];
    const float* dp2_w  = (const float*)d_in[8];
    const float* dp2_b  = (const float*)d_in[9];
    const float* dp3_w  = (const float*)d_in[10];
    const float* dp3_b  = (const float*)d_in[11];
    const float* emb2_w = (const float*)d_in[12];
    const float* emb2_b = (const float*)d_in[13];
    const float* tne_lt = (const float*)d_in[14];
    const float* s0_w   = (const float*)d_in[15];
    const float* s0_b   = (const float*)d_in[16];
    const float* s1_w   = (const float*)d_in[17];
    const float* s1_b   = (const float*)d_in[18];
    const float* ln_g   = (const float*)d_in[19];
    const float* ln_b   = (const float*)d_in[20];
    const float* l0_w   = (const float*)d_in[21];
    const float* l0_b   = (const float*)d_in[22];
    const float* l1_w   = (const float*)d_in[23];
    const float* l1_b   = (const float*)d_in[24];
    const float* l2_w   = (const float*)d_in[25];
    const float* l2_b   = (const float*)d_in[26];
    const float* lx     = (const float*)d_in[27];
    const float* lt     = (const float*)d_in[28];

    float* X  = (float*)d_out;                      // [N,H,9] — X state lives here
    float* ws = (float*)d_ws;

    const size_t EH = (size_t)NEDGE * HC;           // 4,194,304
    const size_t NH = (size_t)NNODE * HC;           //   524,288

    // Workspace layout (floats), regions reused across phases:
    float* Cbuf = ws;                               // E
    float* evn  = ws + NEDGE;                       // 3E
    float* bigA = ws + 4 * (size_t)NEDGE;           // 3*EH : W123 | ea2
    float* bigB = bigA + 3 * EH;                    //   EH : Zij  | ea0
    float* bigC = bigB + EH;                        // 2*EH : Zcat | ea1
    float* acc  = bigC + 2 * EH;                    // 10*NH : scatter accum | X-decomp | AB-decomp
    float* cmx  = acc + 10 * NH;                    // 10*NH : cmix outputs
    float* msg  = cmx + 10 * NH;                    // 10*NH : message accumulators
    float* Yb   = msg + 10 * NH;                    //  9*NH : Y full
    float* sm   = Yb + 9 * NH;                      //  6*NH : normln | h1 | fsc
    float* normln = sm;
    float* h1   = sm + NH;
    float* fsc  = sm + 3 * NH;

    auto gemm = [&](const float* A, int lda, int ka, const float* B,
                    const float* bias, const float* rs, float* O, int ldo, int ko,
                    int M, int Nc, int K, int Z, int za, int zo, int act) {
        int tiles = (M / 64) * (Nc / 16);
        dim3 g((tiles + 7) / 8, 1, Z);
        if (ka == 1)
            k_gemm_wmma<1><<<g, dim3(256), 0, stream>>>(A, lda, B, bias, rs, O, ldo, ko, M, Nc, K, za, zo, act);
        else if (ka == 3)
            k_gemm_wmma<3><<<g, dim3(256), 0, stream>>>(A, lda, B, bias, rs, O, ldo, ko, M, Nc, K, za, zo, act);
        else
            k_gemm_wmma<6><<<g, dim3(256), 0, stream>>>(A, lda, B, bias, rs, O, ldo, ko, M, Nc, K, za, zo, act);
    };
    auto blocks = [](size_t n) { return (unsigned)((n + 255) / 256); };

    // ---------------- TensorNeighborEmbedding ----------------
    k_edge_prep<<<blocks(NEDGE), 256, 0, stream>>>(ew, evec, ei, Cbuf, evn);
    k_zcat<<<blocks((size_t)NEDGE * 2 * HC), 256, 0, stream>>>(z, ei, emb, bigC);
    // Zij = Zcat @ emb2_w + emb2_b
    gemm(bigC, 2 * HC, 1, emb2_w, emb2_b, nullptr, bigB, HC, 1, NEDGE, HC, 2 * HC, 1, 0, 0, 0);
    // W1/W2/W3 = (edge_attr @ dp_w + b) * C
    gemm(eattr, RC, 1, dp1_w, dp1_b, Cbuf, bigA + 0 * EH, HC, 1, NEDGE, HC, RC, 1, 0, 0, 0);
    gemm(eattr, RC, 1, dp2_w, dp2_b, Cbuf, bigA + 1 * EH, HC, 1, NEDGE, HC, RC, 1, 0, 0, 0);
    gemm(eattr, RC, 1, dp3_w, dp3_b, Cbuf, bigA + 2 * EH, HC, 1, NEDGE, HC, RC, 1, 0, 0, 0);
    k_zero<<<blocks(10 * NH), 256, 0, stream>>>(acc, 10 * NH);
    k_tne_scatter<<<blocks(EH), 256, 0, stream>>>(bigB, bigA, evn, ei,
                                                  acc, acc + NH, acc + 4 * NH);
    k_norm_ln<<<NNODE, HC, 0, stream>>>(acc, acc + NH, acc + 4 * NH, ln_g, ln_b, normln);
    // cmix with tne_lt (compact-component strided GEMMs)
    gemm(acc,          HC,     1, tne_lt + 0 * HC * HC, nullptr, nullptr, cmx,          HC,     1, NNODE, HC, HC, 1, 0, 0, 0);
    gemm(acc + NH,     3 * HC, 3, tne_lt + 1 * HC * HC, nullptr, nullptr, cmx + NH,     3 * HC, 3, NNODE, HC, HC, 3, 1, 1, 0);
    gemm(acc + 4 * NH, 6 * HC, 6, tne_lt + 2 * HC * HC, nullptr, nullptr, cmx + 4 * NH, 6 * HC, 6, NNODE, HC, HC, 6, 1, 1, 0);
    // fsc = silu(silu(norm @ s0 + b0) @ s1 + b1)
    gemm(normln, HC, 1, s0_w, s0_b, nullptr, h1, 2 * HC, 1, NNODE, 2 * HC, HC, 1, 0, 0, 1);
    gemm(h1, 2 * HC, 1, s1_w, s1_b, nullptr, fsc, 3 * HC, 1, NNODE, 3 * HC, 2 * HC, 1, 0, 0, 1);
    k_assemble_X<<<blocks(NH), 256, 0, stream>>>(cmx, cmx + NH, cmx + 4 * NH, fsc, X);

    // ---------------- TensorMessage layers ----------------
    for (int l = 0; l < NLAY; ++l) {
        // ea chain: silu MLPs, final scaled by cutoff
        gemm(eattr, RC, 1, l0_w + (size_t)l * RC * HC, l0_b + (size_t)l * HC, nullptr,
             bigB, HC, 1, NEDGE, HC, RC, 1, 0, 0, 1);
        gemm(bigB, HC, 1, l1_w + (size_t)l * HC * 2 * HC, l1_b + (size_t)l * 2 * HC, nullptr,
             bigC, 2 * HC, 1, NEDGE, 2 * HC, HC, 1, 0, 0, 1);
        gemm(bigC, 2 * HC, 1, l2_w + (size_t)l * 2 * HC * 3 * HC, l2_b + (size_t)l * 3 * HC, Cbuf,
             bigA, 3 * HC, 1, NEDGE, 3 * HC, 2 * HC, 1, 0, 0, 1);
        // X <- X/(||X||^2+1); compact decompose
        k_decomp<<<blocks(NH), 256, 0, stream>>>(X, acc, acc + NH, acc + 4 * NH);
        // cmix with msg_lx
        gemm(acc,          HC,     1, lx + (size_t)(l * 3 + 0) * HC * HC, nullptr, nullptr, cmx,          HC,     1, NNODE, HC, HC, 1, 0, 0, 0);
        gemm(acc + NH,     3 * HC, 3, lx + (size_t)(l * 3 + 1) * HC * HC, nullptr, nullptr, cmx + NH,     3 * HC, 3, NNODE, HC, HC, 3, 1, 1, 0);
        gemm(acc + 4 * NH, 6 * HC, 6, lx + (size_t)(l * 3 + 2) * HC * HC, nullptr, nullptr, cmx + 4 * NH, 6 * HC, 6, NNODE, HC, HC, 6, 1, 1, 0);
        k_assembleY<<<blocks(NH), 256, 0, stream>>>(cmx, cmx + NH, cmx + 4 * NH, Yb);
        k_zero<<<blocks(10 * NH), 256, 0, stream>>>(msg, 10 * NH);
        k_msg_scatter<<<blocks(EH), 256, 0, stream>>>(bigA, ei, cmx, cmx + NH, cmx + 4 * NH,
                                                      msg, msg + NH, msg + 4 * NH);
        k_AB<<<blocks(NH), 256, 0, stream>>>(msg, msg + NH, msg + 4 * NH, Yb,
                                             acc, acc + NH, acc + 4 * NH);
        // cmix with msg_lt
        gemm(acc,          HC,     1, lt + (size_t)(l * 3 + 0) * HC * HC, nullptr, nullptr, cmx,          HC,     1, NNODE, HC, HC, 1, 0, 0, 0);
        gemm(acc + NH,     3 * HC, 3, lt + (size_t)(l * 3 + 1) * HC * HC, nullptr, nullptr, cmx + NH,     3 * HC, 3, NNODE, HC, HC, 3, 1, 1, 0);
        gemm(acc + 4 * NH, 6 * HC, 6, lt + (size_t)(l * 3 + 2) * HC * HC, nullptr, nullptr, cmx + 4 * NH, 6 * HC, 6, NNODE, HC, HC, 6, 1, 1, 0);
        k_update<<<blocks(NH), 256, 0, stream>>>(cmx, cmx + NH, cmx + 4 * NH, X);
    }
}